// ExplicitDurationHSMM_1365799600840
// MI455X (gfx1250) — compile-verified
//
#include <hip/hip_runtime.h>

#define TT 8192
#define KK 24
#define UU 104
#define NEGBIG -1.0e18f

typedef __attribute__((ext_vector_type(2))) float v2f;
typedef __attribute__((ext_vector_type(8))) float v8f;

// ---------------------------------------------------------------------------
// Kernel 1: weighted-emission prefix sums.  cum[(t+1)*K+k] = sum_{s<=t} w_s*e[s,k]
// Two-pass chunked scan, single block (8 chunks x 24 states = 192 threads).
// ---------------------------------------------------------------------------
__global__ void hsmm_cumsum(const float* __restrict__ emis,
                            const float* __restrict__ qw,
                            float* __restrict__ cum) {
  __shared__ float part[8 * KK];
  const int tid = threadIdx.x;
  const int k = tid % KK;
  const int c = tid / KK;     // chunk; active if c < 8
  const int CH = TT / 8;      // 1024
  if (c < 8) {
    float s = 0.f;
    const int t0 = c * CH;
    for (int t = t0; t < t0 + CH; ++t)
      s += emis[t * KK + k] * qw[t];
    part[c * KK + k] = s;
  }
  __syncthreads();
  if (c < 8) {
    float run = 0.f;
    for (int cc = 0; cc < c; ++cc) run += part[cc * KK + k];
    if (c == 0) cum[k] = 0.f;
    const int t0 = c * CH;
    for (int t = t0; t < t0 + CH; ++t) {
      run += emis[t * KK + k] * qw[t];
      cum[(t + 1) * KK + k] = run;
    }
  }
}

// ---------------------------------------------------------------------------
// Kernel 2: forward scan.  Single 256-thread block (8 wave32 waves).
// Per step t:
//   A) E[u,:] = exp(alpha[t-1-u,:] - m_u)            (u < min(t,U))
//   B) S = E(112x24) @ A(24x24) via v_wmma_f32_16x16x4_f32 (84 WMMA / step)
//   C) alpha_t[k] = logsumexp_u( trans/log_pi + log_D[k,u] + seg(t,u,k) )
// ---------------------------------------------------------------------------
__global__ void __launch_bounds__(256)
hsmm_forward(const float* __restrict__ cum,
             const float* __restrict__ emis,
             const float* __restrict__ log_pi,
             const float* __restrict__ log_A,
             const int*   __restrict__ tmask,
             const float* __restrict__ log_D,
             float* __restrict__ alphaG) {
  __shared__ float sE[112 * 25];     // exp(prev - m), rows padded 104->112
  __shared__ float sS[112 * 25];     // WMMA result (pre-log trans)
  __shared__ float sRing[UU * 25];   // ring of last U alphas, slot = t % U
  __shared__ float sM[112];          // per-row max shifts
  __shared__ float sA[KK * 32];      // exp(masked log_A), cols padded to 32
  __shared__ float sD[KK * UU];      // log_D[k][u]
  __shared__ float sPi[KK];
  __shared__ float sRedM[KK * 10];
  __shared__ float sRedS[KK * 10];

  const int tid = threadIdx.x;

  for (int i = tid; i < 112 * 25; i += 256) { sE[i] = 0.f; sS[i] = 0.f; }
  for (int i = tid; i < UU * 25; i += 256) sRing[i] = NEGBIG;
  for (int i = tid; i < 112; i += 256) sM[i] = 0.f;
  for (int i = tid; i < KK * 32; i += 256) {
    const int r = i / 32, cc = i % 32;
    float v = 0.f;
    if (cc < KK && tmask[r * KK + cc] != 0) v = __expf(log_A[r * KK + cc]);
    sA[i] = v;
  }
  for (int i = tid; i < KK * UU; i += 256) sD[i] = log_D[i];
  if (tid < KK) {
    sPi[tid] = log_pi[tid];
    const float a0 = log_pi[tid] + log_D[tid * UU] + emis[tid];
    sRing[tid] = a0;                 // slot 0 (= 0 % U), stride-25 row 0
    alphaG[tid] = a0;
  }
  __syncthreads();

  const int lane = tid & 31;
  const int wave = tid >> 5;
  const int h = lane >> 4;           // lane half (K/M split per ISA layout)
  const int lm = lane & 15;
  const int dk = tid % KK;           // reduction: state index
  const int dc = tid / KK;           // reduction: chunk index (< 10 active)

  for (int t = 1; t < TT; ++t) {
    const int umax = (t < UU) ? t : UU;          // rows with a real trans
    const int ulim = (t + 1 < UU) ? (t + 1) : UU; // durations contributing

    // ---- phase A: shift + exponentiate previous alphas -------------------
    if (tid < umax) {
      const int u = tid;
      const int row = (t - 1 - u) % UU;
      float mx = -3.0e38f;
      for (int i = 0; i < KK; ++i) mx = fmaxf(mx, sRing[row * 25 + i]);
      sM[u] = mx;
      for (int i = 0; i < KK; ++i)
        sE[u * 25 + i] = __expf(sRing[row * 25 + i] - mx);
    }
    __syncthreads();

    // ---- phase B: S = E @ A with f32 WMMA (7 M-tiles x 2 N-tiles) --------
    for (int tile = wave; tile < 14; tile += 8) {
      const int mb = (tile % 7) * 16;
      const int nb = (tile / 7) * 16;
      v8f acc = {0.f, 0.f, 0.f, 0.f, 0.f, 0.f, 0.f, 0.f};
#pragma unroll
      for (int kk = 0; kk < KK; kk += 4) {
        v2f a, b;
        // A frag (16x4 f32): lanes 0-15 hold K=kk,kk+1; lanes 16-31 kk+2,kk+3
        a.x = sE[(mb + lm) * 25 + kk + 2 * h];
        a.y = sE[(mb + lm) * 25 + kk + 2 * h + 1];
        // B frag (4x16 f32): same half-split pattern over the K rows
        b.x = sA[(kk + 2 * h) * 32 + nb + lm];
        b.y = sA[(kk + 2 * h + 1) * 32 + nb + lm];
        acc = __builtin_amdgcn_wmma_f32_16x16x4_f32(false, a, false, b,
                                                    (short)0, acc, false, false);
      }
      // C/D layout: VGPR v, lane-half h -> row v + 8h, col lm
#pragma unroll
      for (int vv = 0; vv < 8; ++vv)
        sS[(mb + vv + 8 * h) * 25 + nb + lm] = acc[vv];
    }
    __syncthreads();

    // ---- phase C: alpha_t[k] = logsumexp over durations ------------------
    float lv[11];
    float Mk = -3.0e38f, lmax = -3.0e38f;
    if (dc < 10) {
      const float cumT = cum[(t + 1) * KK + dk];
#pragma unroll
      for (int j = 0; j < 11; ++j) {
        const int u = dc + 10 * j;
        float v = -3.0e38f;
        if (u < ulim) {
          const float tr = (u < umax)
              ? (sM[u] + __logf(fmaxf(sS[u * 25 + dk], 1.0e-30f)))
              : sPi[dk];                       // u == t (< U): entry from pi
          v = tr + sD[dk * UU + u] + (cumT - cum[(t - u) * KK + dk]);
        }
        lv[j] = v;
        lmax = fmaxf(lmax, v);
      }
      sRedM[dk * 10 + dc] = lmax;
    }
    __syncthreads();
    if (dc < 10) {
      for (int cc = 0; cc < 10; ++cc) Mk = fmaxf(Mk, sRedM[dk * 10 + cc]);
      float s = 0.f;
#pragma unroll
      for (int j = 0; j < 11; ++j) s += __expf(lv[j] - Mk);
      sRedS[dk * 10 + dc] = s;
    }
    __syncthreads();
    if (dc == 0) {
      float tot = 0.f;
      for (int cc = 0; cc < 10; ++cc) tot += sRedS[dk * 10 + cc];
      const float at = Mk + __logf(tot);
      sRing[(t % UU) * 25 + dk] = at;
      alphaG[t * KK + dk] = at;
    }
    __syncthreads();
  }
}

// ---------------------------------------------------------------------------
// Kernel 3: backward scan (light: 104x24 adds + 24x24 log-matvec per step).
// ---------------------------------------------------------------------------
__global__ void __launch_bounds__(256)
hsmm_backward(const float* __restrict__ cum,
              const float* __restrict__ log_A,
              const int*   __restrict__ tmask,
              const float* __restrict__ log_D,
              float* __restrict__ betaG) {
  __shared__ float sRing[UU * 25];   // ring of betas, slot = t % U
  __shared__ float sA[KK * KK];      // exp(masked log_A)
  __shared__ float sD[KK * UU];
  __shared__ float sMv[KK];
  __shared__ float sE2[KK];
  __shared__ float sRedM[KK * 10];
  __shared__ float sRedS[KK * 10];

  const int tid = threadIdx.x;
  for (int i = tid; i < UU * 25; i += 256) sRing[i] = NEGBIG;
  for (int i = tid; i < KK * KK; i += 256)
    sA[i] = (tmask[i] != 0) ? __expf(log_A[i]) : 0.f;
  for (int i = tid; i < KK * UU; i += 256) sD[i] = log_D[i];
  if (tid < KK) {
    sRing[((TT - 1) % UU) * 25 + tid] = 0.f;   // beta[T-1] = 0
    betaG[(TT - 1) * KK + tid] = 0.f;
  }
  __syncthreads();

  const int dk = tid % KK;
  const int dc = tid / KK;

  for (int t = TT - 2; t >= 0; --t) {
    const int un = (TT - 1 - t < UU) ? (TT - 1 - t) : UU;

    // m[k] = logsumexp_u( log_D[k,u] + (cum[t+2+u]-cum[t+1]) + beta[t+1+u] )
    float lv[11];
    float Mk = -3.0e38f, lmax = -3.0e38f;
    if (dc < 10) {
      const float cumT1 = cum[(t + 1) * KK + dk];
#pragma unroll
      for (int j = 0; j < 11; ++j) {
        const int u = dc + 10 * j;
        float v = -3.0e38f;
        if (u < un) {
          v = sD[dk * UU + u] + (cum[(t + 2 + u) * KK + dk] - cumT1)
              + sRing[((t + 1 + u) % UU) * 25 + dk];
        }
        lv[j] = v;
        lmax = fmaxf(lmax, v);
      }
      sRedM[dk * 10 + dc] = lmax;
    }
    __syncthreads();
    if (dc < 10) {
      for (int cc = 0; cc < 10; ++cc) Mk = fmaxf(Mk, sRedM[dk * 10 + cc]);
      float s = 0.f;
#pragma unroll
      for (int j = 0; j < 11; ++j) s += __expf(lv[j] - Mk);
      sRedS[dk * 10 + dc] = s;
    }
    __syncthreads();
    if (dc == 0) {
      float tot = 0.f;
      for (int cc = 0; cc < 10; ++cc) tot += sRedS[dk * 10 + cc];
      sMv[dk] = Mk + __logf(fmaxf(tot, 1.0e-37f));
    }
    __syncthreads();
    // beta_t[j] = logsumexp_k( log_A_m[j,k] + m[k] ) as mm + log(A @ exp(m-mm))
    if (tid < KK) {
      float mm = -3.0e38f;
      for (int k = 0; k < KK; ++k) mm = fmaxf(mm, sMv[k]);
      sE2[tid] = __expf(sMv[tid] - mm);
    }
    __syncthreads();
    if (tid < KK) {
      float mm = -3.0e38f;
      for (int k = 0; k < KK; ++k) mm = fmaxf(mm, sMv[k]);
      float s = 0.f;
      for (int k = 0; k < KK; ++k) s += sA[tid * KK + k] * sE2[k];
      const float bt = mm + __logf(fmaxf(s, 1.0e-30f));
      sRing[(t % UU) * 25 + tid] = bt;
      betaG[t * KK + tid] = bt;
    }
    __syncthreads();
  }
}

// ---------------------------------------------------------------------------
// Kernel 4: posterior = softmax(alpha + beta) per row.
// ---------------------------------------------------------------------------
__global__ void hsmm_posterior(const float* __restrict__ alphaG,
                               const float* __restrict__ betaG,
                               float* __restrict__ out) {
  const int r = blockIdx.x * blockDim.x + threadIdx.x;
  if (r >= TT) return;
  float v[KK];
  float mx = -3.0e38f;
#pragma unroll
  for (int k = 0; k < KK; ++k) {
    v[k] = alphaG[r * KK + k] + betaG[r * KK + k];
    mx = fmaxf(mx, v[k]);
  }
  float s = 0.f;
#pragma unroll
  for (int k = 0; k < KK; ++k) { v[k] = __expf(v[k] - mx); s += v[k]; }
  const float inv = 1.0f / s;
#pragma unroll
  for (int k = 0; k < KK; ++k) out[r * KK + k] = v[k] * inv;
}

extern "C" void kernel_launch(void* const* d_in, const int* in_sizes, int n_in,
                              void* d_out, int out_size, void* d_ws, size_t ws_size,
                              hipStream_t stream) {
  const float* emis = (const float*)d_in[0];   // (T,K) f32
  const float* qw   = (const float*)d_in[1];   // (T,)  f32
  const float* lpi  = (const float*)d_in[2];   // (K,)  f32
  const float* lA   = (const float*)d_in[3];   // (K,K) f32
  const float* lD   = (const float*)d_in[4];   // (K,U) f32
  const int*   msk  = (const int*)d_in[5];     // (K,K) i32
  float* out = (float*)d_out;                  // (T,K) f32

  float* cum   = (float*)d_ws;                 // (T+1)*K floats
  float* alpha = cum + ((TT + 1) * KK + 8);    // aligned past cum
  float* beta  = alpha + (TT * KK);            // total ws use ~2.3 MB

  hsmm_cumsum  <<<1, 256, 0, stream>>>(emis, qw, cum);
  hsmm_forward <<<1, 256, 0, stream>>>(cum, emis, lpi, lA, msk, lD, alpha);
  hsmm_backward<<<1, 256, 0, stream>>>(cum, lA, msk, lD, beta);
  hsmm_posterior<<<(TT + 255) / 256, 256, 0, stream>>>(alpha, beta, out);
}